// Model_60705067761776
// MI455X (gfx1250) — compile-verified
//
#include <hip/hip_runtime.h>

typedef __attribute__((ext_vector_type(2))) float v2f;
typedef __attribute__((ext_vector_type(8))) float v8f;

#define D 64

// ---------------------------------------------------------------------------
// Zero fill (float4 granularity) — used to clear agg accumulator + deg counts
// ---------------------------------------------------------------------------
__global__ __launch_bounds__(256) void zero_f32(float* __restrict__ p, int n4) {
  int i = blockIdx.x * blockDim.x + threadIdx.x;
  if (i < n4) ((float4*)p)[i] = make_float4(0.f, 0.f, 0.f, 0.f);
}

// ---------------------------------------------------------------------------
// Scatter-add aggregation: agg[dst] += h[src], deg[dst] += 1
// 16 lanes per edge, each lane moves one float4 (coalesced 256B per edge).
// Atomics land in L2 (agg is 25.6MB, L2 is 192MB -> fully resident).
// ---------------------------------------------------------------------------
__global__ __launch_bounds__(256) void scatter_mean_accum(
    const float* __restrict__ h, const int* __restrict__ src,
    const int* __restrict__ dst, float* __restrict__ agg,
    int* __restrict__ deg, int E) {
  int tid = blockIdx.x * blockDim.x + threadIdx.x;
  int e = tid >> 4;
  int c = tid & 15;
  if (e >= E) return;
  int s = src[e];
  int d = dst[e];
  float4 v = ((const float4*)(h + (size_t)s * D))[c];
  float* outp = agg + (size_t)d * D + c * 4;
  atomicAdd(outp + 0, v.x);
  atomicAdd(outp + 1, v.y);
  atomicAdd(outp + 2, v.z);
  atomicAdd(outp + 3, v.w);
  if (c == 0) atomicAdd(deg + d, 1);
}

// ---------------------------------------------------------------------------
// One K-sweep of out += A(16x64) @ W(64x64)^T for a 16-row node tile.
// V_WMMA_F32_16X16X4_F32 fragments (wave32):
//   A: lane l -> row l&15, K = 4*kstep + 2*(l>>4) + {0,1}
//   B: lane l -> col l&15, same K split (B[k][n] = W[n][k])
//   C/D: vgpr r -> row r + 8*(l>>4), col l&15
// Four independent accumulator chains (N-tiles 0..3) for WMMA ILP.
// ---------------------------------------------------------------------------
template <bool SCALE>
__device__ inline void gemm_accum_16x64(const float* __restrict__ ap,
                                        const float* __restrict__ W, float s,
                                        int row, int khalf, v8f& acc0,
                                        v8f& acc1, v8f& acc2, v8f& acc3) {
#pragma unroll
  for (int kstep = 0; kstep < 16; ++kstep) {
    int k0 = kstep * 4 + khalf * 2;
    v2f a;
    a.x = ap[k0];
    a.y = ap[k0 + 1];
    if (SCALE) { a.x *= s; a.y *= s; }
    const float* w = W + k0;
    v2f b0, b1, b2, b3;
    b0.x = w[(row +  0) * D]; b0.y = w[(row +  0) * D + 1];
    b1.x = w[(row + 16) * D]; b1.y = w[(row + 16) * D + 1];
    b2.x = w[(row + 32) * D]; b2.y = w[(row + 32) * D + 1];
    b3.x = w[(row + 48) * D]; b3.y = w[(row + 48) * D + 1];
    acc0 = __builtin_amdgcn_wmma_f32_16x16x4_f32(false, a, false, b0, (short)0, acc0, false, false);
    acc1 = __builtin_amdgcn_wmma_f32_16x16x4_f32(false, a, false, b1, (short)0, acc1, false, false);
    acc2 = __builtin_amdgcn_wmma_f32_16x16x4_f32(false, a, false, b2, (short)0, acc2, false, false);
    acc3 = __builtin_amdgcn_wmma_f32_16x16x4_f32(false, a, false, b3, (short)0, acc3, false, false);
  }
}

// ---------------------------------------------------------------------------
// out = [A1 * (MEAN ? 1/max(deg,1) : 1)] @ W1^T + bias (+ A2 @ W2^T) (+ ReLU)
// One wave per 16-node tile; wave computes the full 16x64 output tile.
// ---------------------------------------------------------------------------
template <bool MEAN, bool SECOND, bool RELU>
__global__ __launch_bounds__(256) void sage_gemm(
    const float* __restrict__ A1, const float* __restrict__ W1,
    const float* __restrict__ bias, const float* __restrict__ A2,
    const float* __restrict__ W2, const int* __restrict__ deg,
    float* __restrict__ out, int ntiles) {
  int wave = blockIdx.x * (blockDim.x >> 5) + (threadIdx.x >> 5);
  if (wave >= ntiles) return;  // wave-uniform: surviving waves have EXEC all-1s
  int lane = threadIdx.x & 31;
  int row = lane & 15;
  int khalf = lane >> 4;
  size_t node = (size_t)wave * 16 + row;

  float s = 1.0f;
  if (MEAN) s = 1.0f / fmaxf((float)deg[node], 1.0f);

  v8f acc0 = {}, acc1 = {}, acc2 = {}, acc3 = {};

  gemm_accum_16x64<MEAN>(A1 + node * D, W1, s, row, khalf, acc0, acc1, acc2, acc3);
  if (SECOND)
    gemm_accum_16x64<false>(A2 + node * D, W2, 1.0f, row, khalf, acc0, acc1, acc2, acc3);

  float bv0 = bias[row +  0];
  float bv1 = bias[row + 16];
  float bv2 = bias[row + 32];
  float bv3 = bias[row + 48];

  size_t obase = (size_t)wave * 16 + (khalf ? 8 : 0);
#pragma unroll
  for (int r = 0; r < 8; ++r) {
    float o0 = acc0[r] + bv0;
    float o1 = acc1[r] + bv1;
    float o2 = acc2[r] + bv2;
    float o3 = acc3[r] + bv3;
    if (RELU) {
      o0 = fmaxf(o0, 0.f); o1 = fmaxf(o1, 0.f);
      o2 = fmaxf(o2, 0.f); o3 = fmaxf(o3, 0.f);
    }
    float* op = out + (obase + r) * D + row;
    op[ 0] = o0;
    op[16] = o1;
    op[32] = o2;
    op[48] = o3;
  }
}

// ---------------------------------------------------------------------------
// Edge classifier: preds[e] = dot(h[a[e]], h[b[e]]); 16 lanes per edge,
// float4 per lane, wave32 xor-shuffle tree reduction within the 16-lane group.
// ---------------------------------------------------------------------------
__global__ __launch_bounds__(256) void edge_dot(const float* __restrict__ h,
                                                const int* __restrict__ ai,
                                                const int* __restrict__ bi,
                                                float* __restrict__ preds,
                                                int E) {
  int tid = blockIdx.x * blockDim.x + threadIdx.x;
  int e = tid >> 4;
  int c = tid & 15;
  if (e >= E) return;
  int a = ai[e];
  int b = bi[e];
  float4 va = ((const float4*)(h + (size_t)a * D))[c];
  float4 vb = ((const float4*)(h + (size_t)b * D))[c];
  float d = va.x * vb.x + va.y * vb.y + va.z * vb.z + va.w * vb.w;
  d += __shfl_xor(d, 1, 32);
  d += __shfl_xor(d, 2, 32);
  d += __shfl_xor(d, 4, 32);
  d += __shfl_xor(d, 8, 32);
  if (c == 0) preds[e] = d;
}

// ---------------------------------------------------------------------------
extern "C" void kernel_launch(void* const* d_in, const int* in_sizes, int n_in,
                              void* d_out, int out_size, void* d_ws,
                              size_t ws_size, hipStream_t stream) {
  const float* x     = (const float*)d_in[0];
  const int*   ei    = (const int*)d_in[1];
  const int*   eli   = (const int*)d_in[2];
  const float* W_lin = (const float*)d_in[3];
  const float* b_lin = (const float*)d_in[4];
  const float* Wl1   = (const float*)d_in[5];
  const float* bl1   = (const float*)d_in[6];
  const float* Wr1   = (const float*)d_in[7];
  const float* Wl2   = (const float*)d_in[8];
  const float* bl2   = (const float*)d_in[9];
  const float* Wr2   = (const float*)d_in[10];

  const int N  = in_sizes[0] / D;
  const int E  = in_sizes[1] / 2;
  const int EL = in_sizes[2] / 2;

  float* h0  = (float*)d_ws;                 // [N, 64]
  float* h1  = h0 + (size_t)N * D;           // [N, 64]
  float* agg = h1 + (size_t)N * D;           // [N, 64]
  int*   deg = (int*)(agg + (size_t)N * D);  // [N]

  const int ntiles = (N + 15) / 16;
  const dim3 blk(256);
  const int gemm_blocks = (ntiles + 7) / 8;     // 8 waves / block
  const int agg4  = (N * D) / 4;
  const int deg4  = (N + 3) / 4;
  const int sc_blocks = ((E * 16) + 255) / 256;
  const int ed_blocks = ((EL * 16) + 255) / 256;

  // h0 = x @ W_lin^T + b_lin
  sage_gemm<false, false, false><<<gemm_blocks, blk, 0, stream>>>(
      x, W_lin, b_lin, nullptr, nullptr, nullptr, h0, ntiles);

  // conv1: h1 = relu(mean_agg(h0) @ Wl1^T + bl1 + h0 @ Wr1^T)
  zero_f32<<<(agg4 + 255) / 256, blk, 0, stream>>>(agg, agg4);
  zero_f32<<<(deg4 + 255) / 256, blk, 0, stream>>>((float*)deg, deg4);
  scatter_mean_accum<<<sc_blocks, blk, 0, stream>>>(h0, ei, ei + E, agg, deg, E);
  sage_gemm<true, true, true><<<gemm_blocks, blk, 0, stream>>>(
      agg, Wl1, bl1, h0, Wr1, deg, h1, ntiles);

  // conv2: h2 (reuse h0) = mean_agg(h1) @ Wl2^T + bl2 + h1 @ Wr2^T
  zero_f32<<<(agg4 + 255) / 256, blk, 0, stream>>>(agg, agg4);
  zero_f32<<<(deg4 + 255) / 256, blk, 0, stream>>>((float*)deg, deg4);
  scatter_mean_accum<<<sc_blocks, blk, 0, stream>>>(h1, ei, ei + E, agg, deg, E);
  sage_gemm<true, true, false><<<gemm_blocks, blk, 0, stream>>>(
      agg, Wl2, bl2, h1, Wr2, deg, h0, ntiles);

  // preds[e] = dot(h2[src], h2[dst])
  edge_dot<<<ed_blocks, blk, 0, stream>>>(h0, eli, eli + EL, (float*)d_out, EL);
}